// SemanticDecoupling_55396488184259
// MI455X (gfx1250) — compile-verified
//
#include <hip/hip_runtime.h>
#include <math.h>

typedef __attribute__((ext_vector_type(2))) float v2f;
typedef __attribute__((ext_vector_type(4))) float f4;
typedef __attribute__((ext_vector_type(8))) float v8f;

#define B_   4
#define C_   2048
#define HW_  196
#define P_   784      // B_*HW_
#define K_   80
#define WD_  300
#define D_   1024

#if defined(__HIP_DEVICE_COMPILE__) && __has_builtin(__builtin_amdgcn_tanhf)
#define TANH_F32(x) __builtin_amdgcn_tanhf(x)
#else
#define TANH_F32(x) tanhf(x)
#endif

// ---------------- stage 1: img_feat (B,C,H,W) -> x (P,C) contiguous ---------
__global__ void transpose_kernel(const float* __restrict__ img, float* __restrict__ x) {
    __shared__ float tile[32][33];
    const int b  = blockIdx.x;          // 0..3
    const int ht = blockIdx.y;          // 0..6   (ceil(196/32))
    const int ct = blockIdx.z;          // 0..63  (2048/32)
    int hw = ht * 32 + threadIdx.x;
    int c  = ct * 32 + threadIdx.y;
    if (hw < HW_)
        tile[threadIdx.y][threadIdx.x] = img[((size_t)(b * C_ + c)) * HW_ + hw];
    __syncthreads();
    int hw2 = ht * 32 + threadIdx.y;
    int c2  = ct * 32 + threadIdx.x;
    if (hw2 < HW_)
        x[((size_t)(b * HW_ + hw2)) * C_ + c2] = tile[threadIdx.x][threadIdx.y];
}

// ---------------- stage 2: wproj[k,d] = word_feat[k,:] . W2[d,:] ------------
__global__ void wproj_kernel(const float* __restrict__ wf, const float* __restrict__ W2,
                             float* __restrict__ wproj) {
    int idx = blockIdx.x * blockDim.x + threadIdx.x;   // K_*D_ = 81920
    if (idx >= K_ * D_) return;
    int k = idx / D_, d = idx % D_;
    const float* a = wf + k * WD_;
    const float* b = W2 + d * WD_;
    float acc = 0.f;
    #pragma unroll 4
    for (int v = 0; v < WD_; ++v) acc += a[v] * b[v];
    wproj[idx] = acc;
}

// ---------------- stage 3: v[d] = W4 . W3[:,d];  c0 = b3.W4 + b4 ------------
__global__ void fuse34_kernel(const float* __restrict__ W3, const float* __restrict__ b3,
                              const float* __restrict__ W4, const float* __restrict__ b4,
                              float* __restrict__ vvec, float* __restrict__ c0) {
    __shared__ float red[D_];
    int d = threadIdx.x;                 // 1024 threads
    float acc = 0.f;
    for (int e = 0; e < D_; ++e) acc += W4[e] * W3[(size_t)e * D_ + d];
    vvec[d] = acc;
    red[d] = b3[d] * W4[d];
    __syncthreads();
    for (int s = D_ / 2; s > 0; s >>= 1) {
        if (d < s) red[d] += red[d + s];
        __syncthreads();
    }
    if (d == 0) c0[0] = red[0] + b4[0];
}

// ---------------- stage 4: fc1 GEMM via V_WMMA_F32_16X16X4_F32 --------------
// iproj[p,n] = sum_c x[p,c] * W1[n,c].  4 waves / block, one 16x16 tile each.
// A 16x4 fp32 fragment: VGPR0 holds K=(0|2), VGPR1 holds K=(1|3) per lane half.
__global__ void fc1_wmma_kernel(const float* __restrict__ x, const float* __restrict__ W1,
                                float* __restrict__ iproj) {
    const int mt   = blockIdx.x % 49;                      // 49*16 = 784 rows
    const int nt   = (blockIdx.x / 49) * 4 + (threadIdx.x >> 5); // 64 col tiles
    const int lane = threadIdx.x & 31;                     // wave32, full EXEC
    const int l15  = lane & 15;
    const int kh   = (lane >> 4) * 2;                      // 0 or 2
    const float* arow = x  + (size_t)(mt * 16 + l15) * C_ + kh;
    const float* brow = W1 + (size_t)(nt * 16 + l15) * C_ + kh;

    v8f acc = {};
    #pragma unroll 4
    for (int kb = 0; kb < C_; kb += 4) {
        float2 af = *(const float2*)(arow + kb);
        float2 bf = *(const float2*)(brow + kb);
        v2f a; a[0] = af.x; a[1] = af.y;
        v2f b; b[0] = bf.x; b[1] = bf.y;
        acc = __builtin_amdgcn_wmma_f32_16x16x4_f32(
                  false, a, false, b, (short)0, acc, false, false);
    }
    // D layout: VGPR r -> M = r + (lane<16 ? 0 : 8), N = lane&15
    const int mbase = mt * 16 + ((lane < 16) ? 0 : 8);
    const int ncol  = nt * 16 + l15;
    #pragma unroll
    for (int r = 0; r < 8; ++r)
        iproj[(size_t)(mbase + r) * D_ + ncol] = acc[r];
}

// ---------------- stage 5: coef[p,k] = sum_d tanh(ip*wp)*v[d] + c0 ----------
__global__ void coef_kernel(const float* __restrict__ iproj, const float* __restrict__ wproj,
                            const float* __restrict__ vvec, const float* __restrict__ c0,
                            float* __restrict__ coef) {
    const int lane = threadIdx.x & 31;
    const int row  = blockIdx.x * (blockDim.x / 32) + (threadIdx.x >> 5);
    if (row >= P_ * K_) return;
    const int p = row / K_, k = row % K_;
    const float* ip = iproj + (size_t)p * D_;
    const float* wp = wproj + (size_t)k * D_;
    float acc = 0.f;
    #pragma unroll 4
    for (int d = lane; d < D_; d += 32)
        acc += TANH_F32(ip[d] * wp[d]) * vvec[d];
    #pragma unroll
    for (int off = 16; off > 0; off >>= 1)
        acc += __shfl_xor(acc, off, 32);
    if (lane == 0) coef[row] = acc + c0[0];
}

// ---------------- stage 6: softmax over 196 positions per (b,k) -------------
__global__ void softmax_kernel(const float* __restrict__ coef, float* __restrict__ sm) {
    __shared__ float sd[256];
    const int b = blockIdx.x / K_, k = blockIdx.x % K_;
    const int t = threadIdx.x;
    float val = (t < HW_) ? coef[(size_t)(b * HW_ + t) * K_ + k] : -__builtin_inff();
    sd[t] = val;
    __syncthreads();
    for (int s = 128; s > 0; s >>= 1) {
        if (t < s) sd[t] = fmaxf(sd[t], sd[t + s]);
        __syncthreads();
    }
    const float m = sd[0];
    __syncthreads();
    float e = (t < HW_) ? __expf(val - m) : 0.f;
    sd[t] = e;
    __syncthreads();
    for (int s = 128; s > 0; s >>= 1) {
        if (t < s) sd[t] += sd[t + s];
        __syncthreads();
    }
    if (t < HW_) sm[(size_t)(b * HW_ + t) * K_ + k] = e / sd[0];
}

// ---------------- stage 7: fwc[p,k,c] = x[p,c]*sm[p,k]  (write-BW bound) ----
// x row staged into LDS via gfx1250 async-to-LDS (inline asm, ASYNCcnt);
// streamed out with non-temporal B128 stores so L2 stays clean for stage 8.
__global__ void fwc_kernel(const float* __restrict__ x, const float* __restrict__ sm,
                           float* __restrict__ fwc) {
    __shared__ __align__(16) float xs[C_];
    const int p   = blockIdx.x;              // 0..783
    const int tid = threadIdx.x;             // 256
#if defined(__gfx1250__)
    {
        const char* gbase = (const char*)(x + (size_t)p * C_);
        unsigned lbase = (unsigned)(size_t)(__attribute__((address_space(3))) char*)xs;
        for (int i = tid; i < C_ / 4; i += 256) {          // 512 x 16B chunks
            const char* gp = gbase + (size_t)i * 16;
            unsigned    lp = lbase + (unsigned)i * 16;
            asm volatile("global_load_async_to_lds_b128 %0, %1, off"
                         :: "v"(lp), "v"(gp) : "memory");
        }
        asm volatile("s_wait_asynccnt 0" ::: "memory");
    }
    __syncthreads();
#else
    for (int c = tid; c < C_; c += 256) xs[c] = x[(size_t)p * C_ + c];
    __syncthreads();
#endif
    const float* smrow = sm + (size_t)p * K_;
    const f4* xsv = (const f4*)xs;
    for (int k = 0; k < K_; ++k) {
        const float s = smrow[k];
        f4* outv = (f4*)(fwc + ((size_t)p * K_ + k) * C_);
        #pragma unroll
        for (int i = tid; i < C_ / 4; i += 256) {
            f4 t = xsv[i] * s;
            __builtin_nontemporal_store(t, outv + i);   // TH=NT: bypass L2 residency
        }
    }
}

// ---------------- stage 8: semantic[b,k,c] = sum_hw sm[b,hw,k]*x[b,hw,c] ----
__global__ void semantic_kernel(const float* __restrict__ x, const float* __restrict__ sm,
                                float* __restrict__ sem) {
    const int b = blockIdx.x / K_, k = blockIdx.x % K_;
    const int tid = threadIdx.x;             // 256; 8 channels per thread
    float acc[8] = {0.f, 0.f, 0.f, 0.f, 0.f, 0.f, 0.f, 0.f};
    for (int hw = 0; hw < HW_; ++hw) {
        const float s = sm[(size_t)(b * HW_ + hw) * K_ + k];
        const float* xr = x + (size_t)(b * HW_ + hw) * C_;
        #pragma unroll
        for (int j = 0; j < 8; ++j) acc[j] += s * xr[tid + j * 256];
    }
    float* o = sem + (size_t)(b * K_ + k) * C_;
    #pragma unroll
    for (int j = 0; j < 8; ++j) o[tid + j * 256] = acc[j];
}

extern "C" void kernel_launch(void* const* d_in, const int* in_sizes, int n_in,
                              void* d_out, int out_size, void* d_ws, size_t ws_size,
                              hipStream_t stream) {
    const float* img = (const float*)d_in[0];   // (4,2048,14,14)
    const float* wf  = (const float*)d_in[1];   // (80,300)
    const float* W1  = (const float*)d_in[2];   // (1024,2048)
    const float* W2  = (const float*)d_in[3];   // (1024,300)
    const float* W3  = (const float*)d_in[4];   // (1024,1024)
    const float* b3  = (const float*)d_in[5];   // (1024,)
    const float* W4  = (const float*)d_in[6];   // (1,1024)
    const float* b4  = (const float*)d_in[7];   // (1,)

    // workspace layout (floats), ~9.8 MB total
    float* ws    = (float*)d_ws;
    float* x     = ws;                              // 784*2048   = 1,605,632
    float* iproj = x     + (size_t)P_ * C_;         // 784*1024   =   802,816
    float* wproj = iproj + (size_t)P_ * D_;         // 80*1024    =    81,920
    float* vvec  = wproj + (size_t)K_ * D_;         // 1024
    float* c0    = vvec  + D_;                      // 1 (+pad)
    float* coef  = c0    + 32;                      // 784*80     =    62,720

    // output layout: semantic | fwc | sm
    float* sem = (float*)d_out;
    float* fwc = sem + (size_t)B_ * K_ * C_;                 // +655,360
    float* sm  = fwc + (size_t)P_ * K_ * C_;                 // +128,450,560

    transpose_kernel<<<dim3(B_, 7, C_ / 32), dim3(32, 32), 0, stream>>>(img, x);
    wproj_kernel<<<(K_ * D_ + 255) / 256, 256, 0, stream>>>(wf, W2, wproj);
    fuse34_kernel<<<1, D_, 0, stream>>>(W3, b3, W4, b4, vvec, c0);
    fc1_wmma_kernel<<<49 * (D_ / 16 / 4), 128, 0, stream>>>(x, W1, iproj);
    coef_kernel<<<(P_ * K_) / 8, 256, 0, stream>>>(iproj, wproj, vvec, c0, coef);
    softmax_kernel<<<B_ * K_, 256, 0, stream>>>(coef, sm);
    fwc_kernel<<<P_, 256, 0, stream>>>(x, sm, fwc);
    semantic_kernel<<<B_ * K_, 256, 0, stream>>>(x, sm, sem);
}